// mem_update_66477503807492
// MI455X (gfx1250) — compile-verified
//
#include <hip/hip_runtime.h>

// LIF scan forward:
//   mem = mem * 0.25 * (1 - spike) + x[t];  spike = (mem >= 0.5);  out[t] = spike
// Bandwidth-bound streaming recurrence: 268 MB of traffic, ~11.5 us floor @ 23.3 TB/s.
// CDNA5 path used: GLOBAL_LOAD_ASYNC_TO_LDS_B128 (ASYNCcnt) staging + wave32 b128 stores.

typedef float v4f __attribute__((ext_vector_type(4)));

#define T_STEPS 8
#define BLOCK 256

__global__ __launch_bounds__(BLOCK) void lif_scan_vec_kernel(
    const float* __restrict__ x, float* __restrict__ out,
    long long n_per_t, long long nvec)
{
    // 8 timesteps * 256 lanes * 16B = 32 KB LDS per block (320 KB per WGP available)
    __shared__ v4f tile[T_STEPS][BLOCK];

    const int tid = threadIdx.x;
    const long long v = (long long)blockIdx.x * BLOCK + tid;
    if (v >= nvec) return;
    const long long n0 = v * 4;

    // Stage all 8 timestep tiles into LDS with CDNA5 async global->LDS copies.
    // Each lane copies (and later reads back) only its own 16B slot, so no
    // barrier is needed; s_wait_asynccnt 0 is the only ordering point.
#pragma unroll
    for (int t = 0; t < T_STEPS; ++t) {
        unsigned long long gaddr =
            (unsigned long long)(x + (long long)t * n_per_t + n0);
        // Low 32 bits of a flat shared-aperture address are the LDS byte offset
        // (ISA 10.2).
        unsigned loff = (unsigned)(unsigned long long)&tile[t][tid];
        asm volatile("global_load_async_to_lds_b128 %0, %1, off"
                     :
                     : "v"(loff), "v"(gaddr)
                     : "memory");
    }
    asm volatile("s_wait_asynccnt 0x0" ::: "memory");

    float m0 = 0.f, m1 = 0.f, m2 = 0.f, m3 = 0.f;
    float s0 = 0.f, s1 = 0.f, s2 = 0.f, s3 = 0.f;
#pragma unroll
    for (int t = 0; t < T_STEPS; ++t) {
        v4f xv = tile[t][tid];  // ds_load_b128
        m0 = m0 * (0.25f * (1.0f - s0)) + xv.x; s0 = (m0 >= 0.5f) ? 1.0f : 0.0f;
        m1 = m1 * (0.25f * (1.0f - s1)) + xv.y; s1 = (m1 >= 0.5f) ? 1.0f : 0.0f;
        m2 = m2 * (0.25f * (1.0f - s2)) + xv.z; s2 = (m2 >= 0.5f) ? 1.0f : 0.0f;
        m3 = m3 * (0.25f * (1.0f - s3)) + xv.w; s3 = (m3 >= 0.5f) ? 1.0f : 0.0f;
        v4f sv = { s0, s1, s2, s3 };
        // Write-once output stream: bypass cache rinse with a non-temporal b128 store.
        __builtin_nontemporal_store(sv, (v4f*)(out + (long long)t * n_per_t + n0));
    }
}

// Scalar tail for n_per_t not divisible by 4 (not hit for the reference shape,
// but keeps the launch correct for any size).
__global__ __launch_bounds__(BLOCK) void lif_scan_tail_kernel(
    const float* __restrict__ x, float* __restrict__ out,
    long long n_per_t, long long start)
{
    long long n = start + (long long)blockIdx.x * BLOCK + threadIdx.x;
    if (n >= n_per_t) return;
    float m = 0.f, s = 0.f;
#pragma unroll
    for (int t = 0; t < T_STEPS; ++t) {
        float xv = x[(long long)t * n_per_t + n];
        m = m * (0.25f * (1.0f - s)) + xv;
        s = (m >= 0.5f) ? 1.0f : 0.0f;
        out[(long long)t * n_per_t + n] = s;
    }
}

extern "C" void kernel_launch(void* const* d_in, const int* in_sizes, int n_in,
                              void* d_out, int out_size, void* d_ws, size_t ws_size,
                              hipStream_t stream)
{
    (void)n_in; (void)d_ws; (void)ws_size; (void)out_size;
    const float* x = (const float*)d_in[0];
    float* out = (float*)d_out;

    const long long total   = (long long)in_sizes[0];          // 8*32*128*32*32
    const long long n_per_t = total / T_STEPS;                 // neurons per timestep
    const long long nvec    = n_per_t / 4;                     // float4 work-items

    if (nvec > 0) {
        int blocks = (int)((nvec + BLOCK - 1) / BLOCK);        // 4096 for ref shape
        lif_scan_vec_kernel<<<blocks, BLOCK, 0, stream>>>(x, out, n_per_t, nvec);
    }
    const long long tail_start = nvec * 4;
    const long long tail = n_per_t - tail_start;
    if (tail > 0) {
        int blocks = (int)((tail + BLOCK - 1) / BLOCK);
        lif_scan_tail_kernel<<<blocks, BLOCK, 0, stream>>>(x, out, n_per_t, tail_start);
    }
}